// DiffeqSolver_71012989272606
// MI455X (gfx1250) — compile-verified
//
#include <hip/hip_runtime.h>
#include <math.h>

typedef __attribute__((ext_vector_type(2))) float v2f;
typedef __attribute__((ext_vector_type(8))) float v8f;

#define N_TRAJ   8192
#define N_TIME   64
#define LATENTS  64
#define HIDDEN   128
#define WPB      8      // waves per block (256 threads)
#define TC       8      // time steps per block in main kernel

// Hardware V_TANH_F32 if the toolchain exposes it (CDNA5 TRANS op),
// else fall back to OCML tanhf.
__device__ __forceinline__ float fast_tanh(float x) {
#if __has_builtin(__builtin_amdgcn_tanhf)
  return __builtin_amdgcn_tanhf(x);
#else
  return tanhf(x);
#endif
}

// ---------------------------------------------------------------------------
// Kernel M: active[traj,time] = (sum_d mask[traj,time,d] > 0) ? 1.0f : 0.0f
// 16 lanes cooperate on one (traj,time) pair: each loads int4 (coalesced),
// then a 16-wide xor-shuffle reduction.
// ---------------------------------------------------------------------------
__global__ __launch_bounds__(256) void mask_active_kernel(
    const int* __restrict__ mask, float* __restrict__ actf)
{
  int gid   = blockIdx.x * 256 + threadIdx.x;
  int pair  = gid >> 4;          // 0 .. 8192*64-1
  int chunk = gid & 15;          // 16 int4 chunks of 64 ints
  int4 v = ((const int4*)mask)[(size_t)pair * 16 + chunk];
  int s = v.x + v.y + v.z + v.w;
  s += __shfl_xor(s, 1, 16);
  s += __shfl_xor(s, 2, 16);
  s += __shfl_xor(s, 4, 16);
  s += __shfl_xor(s, 8, 16);
  if (chunk == 0) actf[pair] = (s > 0) ? 1.0f : 0.0f;
}

// ---------------------------------------------------------------------------
// Generic 16-row-tile fp32 WMMA GEMM: Out1[rows][N] = Ain[rows][K] @ W[K][N] (+bias)
// Optionally also Out2 = tanh(Out1). One wave per 16-row tile, WPB waves/block.
// W is staged in LDS pre-swizzled into V_WMMA_F32_16X16X4_F32 B-fragment layout.
// ---------------------------------------------------------------------------
template<int K, int N, bool BIAS, bool TANH_OUT2>
__global__ __launch_bounds__(256) void gemm16_kernel(
    const float* __restrict__ Ain, const float* __restrict__ W,
    const float* __restrict__ bias, float* __restrict__ Out1,
    float* __restrict__ Out2)
{
  constexpr int KS = K / 4;     // k-steps of 4
  constexpr int NT = N / 16;    // 16-wide n-tiles
  __shared__ float lw[K * N];   // 32 KB for both instantiations

  // Stage W[K][N] (row-major, r = K index) into B-fragment layout:
  // frag(k,nt): lanes 0-15 hold (K=4k+{0,1}, N=l); lanes 16-31 (K=4k+{2,3}, N=l)
  for (int idx = threadIdx.x; idx < K * N; idx += 256) {
    int r = idx / N, c = idx % N;
    int k = r >> 2, j = r & 1, half = (r >> 1) & 1;
    int nt = c >> 4, cl = c & 15;
    lw[(k * NT + nt) * 64 + half * 32 + cl * 2 + j] = W[idx];
  }
  __syncthreads();

  int lane = threadIdx.x & 31, wv = threadIdx.x >> 5;
  int l = lane & 15, hi = lane >> 4;
  int tt = (blockIdx.x * WPB + wv) * 16;   // base trajectory row

  // A-fragments: per lane one contiguous f32 pair per k-step -> global b64 loads
  v2f af[KS];
#pragma unroll
  for (int k = 0; k < KS; ++k)
    af[k] = *(const v2f*)(Ain + (size_t)(tt + l) * K + 4 * k + 2 * hi);

  v8f acc[NT] = {};
#pragma unroll
  for (int k = 0; k < KS; ++k) {
#pragma unroll
    for (int nt = 0; nt < NT; ++nt) {
      v2f bf = *(const v2f*)(&lw[(k * NT + nt) * 64 + hi * 32 + l * 2]);
      acc[nt] = __builtin_amdgcn_wmma_f32_16x16x4_f32(
          false, af[k], false, bf, (short)0, acc[nt], false, false);
    }
  }

  // Epilogue: C/D layout -> row M = r + 8*hi, col N = nt*16 + l
#pragma unroll
  for (int nt = 0; nt < NT; ++nt) {
    float bv = BIAS ? bias[nt * 16 + l] : 0.0f;
#pragma unroll
    for (int r = 0; r < 8; ++r) {
      int row = tt + r + 8 * hi;
      float v = acc[nt][r] + bv;
      Out1[(size_t)row * N + nt * 16 + l] = v;
      if (TANH_OUT2) Out2[(size_t)row * N + nt * 16 + l] = fast_tanh(v);
    }
  }
}

// ---------------------------------------------------------------------------
// Kernel B: per (16-traj tile, time i):
//   Hpre = A + h_i*Bv ; D = tanh(Hpre) @ W2 ; y2 = y0 + h_i*f0 + h_i*(D + b2)
//   out = actf * (t>0 ? y2 : y0)
// A,Bv cached in VGPRs as A-fragment pairs; W2 resident in LDS/VGPRs; TC times/block.
// ---------------------------------------------------------------------------
__global__ __launch_bounds__(256) void diffeq_main_kernel(
    const float* __restrict__ A_ws, const float* __restrict__ Bv_ws,
    const float* __restrict__ f0_ws, const float* __restrict__ y0,
    const float* __restrict__ ts, const float* __restrict__ actf,
    const float* __restrict__ W2, const float* __restrict__ b2,
    float* __restrict__ out)
{
  __shared__ float lw[HIDDEN * LATENTS];   // 32 KB, W2 in B-frag layout, NT=4
  for (int idx = threadIdx.x; idx < HIDDEN * LATENTS; idx += 256) {
    int r = idx >> 6, c = idx & 63;
    int k = r >> 2, j = r & 1, half = (r >> 1) & 1;
    int nt = c >> 4, cl = c & 15;
    lw[(k * 4 + nt) * 64 + half * 32 + cl * 2 + j] = W2[idx];
  }
  __syncthreads();

  int lane = threadIdx.x & 31, wv = threadIdx.x >> 5;
  int l = lane & 15, hi = lane >> 4;
  int tt = (blockIdx.x * WPB + wv) * 16;

  // Cache A, Bv as A-fragment pairs (64 floats each per lane)
  v2f Ar[32], Br[32];
#pragma unroll
  for (int k = 0; k < 32; ++k) {
    Ar[k] = *(const v2f*)(A_ws  + (size_t)(tt + l) * HIDDEN + 4 * k + 2 * hi);
    Br[k] = *(const v2f*)(Bv_ws + (size_t)(tt + l) * HIDDEN + 4 * k + 2 * hi);
  }

  // y0, f0, b2 in C/D layout for the epilogue
  float y0v[4][8], f0v[4][8], b2v[4];
#pragma unroll
  for (int nt = 0; nt < 4; ++nt) {
    b2v[nt] = b2[nt * 16 + l];
#pragma unroll
    for (int r = 0; r < 8; ++r) {
      int row = tt + r + 8 * hi;
      y0v[nt][r] = y0[(size_t)row * LATENTS + nt * 16 + l];
      f0v[nt][r] = f0_ws[(size_t)row * LATENTS + nt * 16 + l];
    }
  }

  int i0 = blockIdx.y * TC;
  for (int it = 0; it < TC; ++it) {
    int i = i0 + it;
    float t = ts[i];
    float h = 0.5f * t;

    v8f acc[4] = {};
#pragma unroll
    for (int k = 0; k < 32; ++k) {
      v2f af;
      af.x = fast_tanh(Ar[k].x + h * Br[k].x);
      af.y = fast_tanh(Ar[k].y + h * Br[k].y);
#pragma unroll
      for (int nt = 0; nt < 4; ++nt) {
        v2f bf = *(const v2f*)(&lw[(k * 4 + nt) * 64 + hi * 32 + l * 2]);
        acc[nt] = __builtin_amdgcn_wmma_f32_16x16x4_f32(
            false, af, false, bf, (short)0, acc[nt], false, false);
      }
    }

    float av[8];
#pragma unroll
    for (int r = 0; r < 8; ++r)
      av[r] = actf[(size_t)(tt + r + 8 * hi) * N_TIME + i];

    bool pos = t > 0.0f;
#pragma unroll
    for (int nt = 0; nt < 4; ++nt) {
#pragma unroll
      for (int r = 0; r < 8; ++r) {
        int row = tt + r + 8 * hi;
        float y2  = y0v[nt][r] + h * f0v[nt][r] + h * (acc[nt][r] + b2v[nt]);
        float val = pos ? y2 : y0v[nt][r];
        out[((size_t)row * N_TIME + i) * LATENTS + nt * 16 + l] = av[r] * val;
      }
    }
  }
}

// ---------------------------------------------------------------------------
extern "C" void kernel_launch(void* const* d_in, const int* in_sizes, int n_in,
                              void* d_out, int out_size, void* d_ws, size_t ws_size,
                              hipStream_t stream) {
  (void)in_sizes; (void)n_in; (void)out_size; (void)ws_size;
  const float* y0   = (const float*)d_in[0];   // [8192][64]
  const float* ts   = (const float*)d_in[1];   // [64]
  const int*   mask = (const int*)  d_in[2];   // [8192][64][64]
  const float* W1   = (const float*)d_in[3];   // [64][128]
  const float* b1   = (const float*)d_in[4];   // [128]
  const float* W2   = (const float*)d_in[5];   // [128][64]
  const float* b2   = (const float*)d_in[6];   // [64]
  float* out = (float*)d_out;                  // [8192][64][1][64]

  char* ws = (char*)d_ws;
  float* actf  = (float*)(ws);                 // 2 MB: [8192][64]
  float* A_ws  = (float*)(ws + (2u  << 20));   // 4 MB: [8192][128]
  float* Bv_ws = (float*)(ws + (6u  << 20));   // 4 MB: [8192][128]
  float* H0_ws = (float*)(ws + (10u << 20));   // 4 MB: [8192][128]
  float* f0_ws = (float*)(ws + (14u << 20));   // 2 MB: [8192][64]

  // Active mask reduction (reads 134 MB, dominant input traffic, coalesced)
  mask_active_kernel<<<(N_TRAJ * N_TIME * 16) / 256, 256, 0, stream>>>(mask, actf);

  // Per-trajectory prep GEMMs (64 blocks x 8 waves x 16 trajs = 8192)
  gemm16_kernel<64, 128, true,  true ><<<N_TRAJ / (WPB * 16), 256, 0, stream>>>(
      y0, W1, b1, A_ws, H0_ws);                               // A = y0@W1+b1 ; H0 = tanh(A)
  gemm16_kernel<128, 64, true,  false><<<N_TRAJ / (WPB * 16), 256, 0, stream>>>(
      H0_ws, W2, b2, f0_ws, nullptr);                         // f0 = H0@W2+b2
  gemm16_kernel<64, 128, false, false><<<N_TRAJ / (WPB * 16), 256, 0, stream>>>(
      f0_ws, W1, nullptr, Bv_ws, nullptr);                    // Bv = f0@W1

  // Main fused second-step kernel: 64 traj-groups x 8 time-chunks = 512 blocks
  diffeq_main_kernel<<<dim3(N_TRAJ / (WPB * 16), N_TIME / TC), 256, 0, stream>>>(
      A_ws, Bv_ws, f0_ws, y0, ts, actf, W2, b2, out);
}